// BlockDropout_65412351918511
// MI455X (gfx1250) — compile-verified
//
#include <hip/hip_runtime.h>

// BlockDropout on MI455X (gfx1250).
// out[b,m,d] = mask[b,d] * z[b,m,d];  mask = (noise < 0.8), with per-row
// fallback patch. Memory-bound: ~269 MB/launch -> ~11.5us at 23.3 TB/s.
// Strategy: b128 vector memops everywhere, mask row staged to LDS via the
// gfx1250 async-to-LDS path (ASYNCcnt), NT stores on the output so z stays
// resident in the 192MB L2 across graph replays.

typedef float v4f __attribute__((ext_vector_type(4)));

#define B_   128
#define M_   256
#define D_   1024
#define KEEP_THRESH 0.8f
#define MCHUNK 8                 // M-rows per block in apply kernel
#define CHUNKS_PER_B (M_ / MCHUNK)

// ---- gfx1250 async global->LDS copy (16B per lane), via inline asm so it
// ---- assembles on both ROCm 7.2 and amdgpu-toolchain.
__device__ __forceinline__ void async_copy_b128_to_lds(const float* gsrc, float* lds_dst) {
    // Generic pointer to __shared__ has the LDS byte offset in its low 32 bits
    // VDST of the async op takes that offset.
    unsigned lds_off = (unsigned)(unsigned long long)lds_dst;
    asm volatile("global_load_async_to_lds_b128 %0, %1, off"
                 :
                 : "v"(lds_off), "v"(gsrc)
                 : "memory");
}

__device__ __forceinline__ void wait_asynccnt_zero() {
#if defined(__has_builtin) && __has_builtin(__builtin_amdgcn_s_wait_asynccnt)
    __builtin_amdgcn_s_wait_asynccnt(0);
#else
    asm volatile("s_wait_asynccnt 0" ::: "memory");
#endif
}

// ---------------------------------------------------------------------------
// Kernel 1: build mask[b, 0..D) in d_ws. One block per batch row.
// Reduction happens BEFORE the store so the fallback patch is applied by the
// owning thread itself (no cross-thread store ordering hazard).
// ---------------------------------------------------------------------------
__global__ __launch_bounds__(256) void bd_mask_build(const float* __restrict__ noise,
                                                     const int* __restrict__ fallback_idx,
                                                     float* __restrict__ mask) {
    __shared__ int s_any;
    const int b = blockIdx.x;
    const int t = threadIdx.x;          // 0..255, covers D/4 = 256 float4s

    if (t == 0) s_any = 0;
    __syncthreads();

    const v4f n4 = ((const v4f*)(noise + (size_t)b * D_))[t];
    v4f m4;
    m4.x = (n4.x < KEEP_THRESH) ? 1.0f : 0.0f;
    m4.y = (n4.y < KEEP_THRESH) ? 1.0f : 0.0f;
    m4.z = (n4.z < KEEP_THRESH) ? 1.0f : 0.0f;
    m4.w = (n4.w < KEEP_THRESH) ? 1.0f : 0.0f;

    if ((m4.x + m4.y + m4.z + m4.w) > 0.0f) atomicOr(&s_any, 1);
    __syncthreads();

    if (s_any == 0) {
        // Entire row dropped: force the fallback element to 1.
        const unsigned idx = ((unsigned)fallback_idx[b]) & (D_ - 1);
        if ((int)(idx >> 2) == t) m4[idx & 3u] = 1.0f;
    }
    ((v4f*)(mask + (size_t)b * D_))[t] = m4;
}

// ---------------------------------------------------------------------------
// Kernel 2: out = mask (broadcast over M) * z.
// Block handles (b, 8 consecutive m-rows). The 4KB mask row is staged to LDS
// with the async-to-LDS engine, then each thread streams 8 x b128 of z.
// ---------------------------------------------------------------------------
__global__ __launch_bounds__(256) void bd_apply(const float* __restrict__ z,
                                                const float* __restrict__ mask,
                                                float* __restrict__ out) {
    __shared__ __align__(16) float s_mask[D_];

    const int t     = threadIdx.x;                       // 0..255
    const int b     = blockIdx.x / CHUNKS_PER_B;
    const int mbase = (blockIdx.x % CHUNKS_PER_B) * MCHUNK;

    // Async-stage this batch's mask row into LDS (16B per lane = 4KB total).
    async_copy_b128_to_lds(mask + (size_t)b * D_ + (size_t)t * 4, &s_mask[t * 4]);
    wait_asynccnt_zero();
    __syncthreads();

    const v4f m4 = ((const v4f*)s_mask)[t];              // ds_load_b128

    const size_t base = ((size_t)b * M_ + (size_t)mbase) * D_ + (size_t)t * 4;
#pragma unroll
    for (int r = 0; r < MCHUNK; ++r) {
        const size_t off = base + (size_t)r * D_;
        const v4f zv = *(const v4f*)(z + off);           // RT load: keep z hot in L2
        const v4f ov = zv * m4;
        __builtin_nontemporal_store(ov, (v4f*)(out + off)); // NT store: don't pollute L2
    }
}

// ---------------------------------------------------------------------------
extern "C" void kernel_launch(void* const* d_in, const int* in_sizes, int n_in,
                              void* d_out, int out_size, void* d_ws, size_t ws_size,
                              hipStream_t stream) {
    const float* z     = (const float*)d_in[0];   // [B, M, D] f32
    const float* noise = (const float*)d_in[1];   // [B, D]    f32
    const int*   fidx  = (const int*)d_in[2];     // [B]       int
    float*       out   = (float*)d_out;           // [B, M, D] f32
    float*       mask  = (float*)d_ws;            // [B, D]    f32 scratch (512 KB)

    (void)in_sizes; (void)n_in; (void)out_size; (void)ws_size;

    bd_mask_build<<<dim3(B_), dim3(256), 0, stream>>>(noise, fidx, mask);
    bd_apply<<<dim3(B_ * CHUNKS_PER_B), dim3(256), 0, stream>>>(z, mask, out);
}